// LowPassFilterLayer_28054726377562
// MI455X (gfx1250) — compile-verified
//
#include <hip/hip_runtime.h>

// LowPassFilterLayer (GCN layer): out = relu((P @ X) @ W)
//   P: sparse COO (edge_row, edge_col, edge_val), E = 3.2M, N = 100000 nodes
//   X: [N, 256] fp32, W: [256, 256] fp32
//
// Strategy (MI455X / gfx1250):
//  - support (= P@X) is 102.4 MB -> fits in the 192 MB L2. Scatter with
//    hardware global_atomic_add_f32 (L2 atomic units), gathers of x rows are
//    L2 hits (avg degree 32).
//  - dense GEMM uses V_WMMA_F32_16X16X4_F32 (full fp32 precision matrix op),
//    one wave32 per 16x16 output tile, K-loop of 64 WMMAs, ReLU fused at store.

typedef __attribute__((ext_vector_type(2))) float v2f;
typedef __attribute__((ext_vector_type(8))) float v8f;

#define D_FEAT 256

// ---------------------------------------------------------------- zero scratch
__global__ void lpf_zero_f4(float4* __restrict__ p, long n4) {
  long i = (long)blockIdx.x * blockDim.x + threadIdx.x;
  const long stride = (long)gridDim.x * blockDim.x;
  const float4 z = make_float4(0.f, 0.f, 0.f, 0.f);
  for (; i < n4; i += stride) p[i] = z;
}

// ------------------------------------------------- SpMM: S[row,:] += v * X[col,:]
// One wave32 per edge (grid-stride over edges). Each lane handles 8 floats as
// two float4 gathers + 8 hardware float atomics into the L2-resident buffer.
__global__ void lpf_spmm_scatter(const float* __restrict__ x,
                                 const int* __restrict__ erow,
                                 const int* __restrict__ ecol,
                                 const float* __restrict__ eval,
                                 float* __restrict__ S, int E) {
  const int lane = threadIdx.x & 31;
  int warp = (int)((blockIdx.x * blockDim.x + threadIdx.x) >> 5);
  const int nwarps = (int)((gridDim.x * blockDim.x) >> 5);
  for (int e = warp; e < E; e += nwarps) {
    const int   r = erow[e];
    const int   c = ecol[e];
    const float v = eval[e];
    const float* __restrict__ xr = x + (size_t)c * D_FEAT;
    float* __restrict__ sr = S + (size_t)r * D_FEAT;

    const float4 g0 = *(const float4*)(xr + lane * 4);
    const float4 g1 = *(const float4*)(xr + 128 + lane * 4);

    float* d0 = sr + lane * 4;
    float* d1 = sr + 128 + lane * 4;
    unsafeAtomicAdd(d0 + 0, v * g0.x);
    unsafeAtomicAdd(d0 + 1, v * g0.y);
    unsafeAtomicAdd(d0 + 2, v * g0.z);
    unsafeAtomicAdd(d0 + 3, v * g0.w);
    unsafeAtomicAdd(d1 + 0, v * g1.x);
    unsafeAtomicAdd(d1 + 1, v * g1.y);
    unsafeAtomicAdd(d1 + 2, v * g1.z);
    unsafeAtomicAdd(d1 + 3, v * g1.w);
  }
}

// ------------------------------------- GEMM: out = relu(S @ W), fp32 WMMA path
// One wave32 computes one 16x16 tile of out. K=256 -> 64 x V_WMMA_F32_16X16X4_F32.
// A (16x4 f32) layout: lanes 0-15 = M rows, half=lane>>4 selects K pair
//   (VGPR0,VGPR1) = (A[m][k0+2*half], A[m][k0+2*half+1])
// B (4x16 f32) symmetric: (B[k0+2*half][n], B[k0+2*half+1][n]) per lane.
// C/D (16x16 f32, 8 VGPRs): VGPR v = row v (lanes 0-15) / row v+8 (lanes 16-31).
__global__ void lpf_gemm_relu_wmma(const float* __restrict__ S,
                                   const float* __restrict__ W,
                                   float* __restrict__ out, int M) {
  const int lane = threadIdx.x & 31;
  const int waveInBlock = threadIdx.x >> 5;
  const int wavesPerBlock = blockDim.x >> 5;
  const int tilesN = D_FEAT / 16;                       // 16
  const int tile = blockIdx.x * wavesPerBlock + waveInBlock;
  const int tm = tile / tilesN;
  const int tn = tile - tm * tilesN;
  if (tm * 16 >= M) return;

  const int half = lane >> 4;                           // 0 | 1
  const int l = lane & 15;                              // 0..15
  const float* __restrict__ Arow = S + (size_t)(tm * 16 + l) * D_FEAT;
  const int n = tn * 16 + l;

  v8f c = {};
#pragma unroll 4
  for (int k0 = 0; k0 < D_FEAT; k0 += 4) {
    const int ka = k0 + half * 2;
    v2f a = *(const v2f*)(Arow + ka);
    v2f b;
    b.x = W[ka * D_FEAT + n];
    b.y = W[(ka + 1) * D_FEAT + n];
    // 8 args: (neg_a, A, neg_b, B, c_mod, C, reuse_a, reuse_b)
    c = __builtin_amdgcn_wmma_f32_16x16x4_f32(false, a, false, b,
                                              (short)0, c, false, false);
  }

  float* __restrict__ O = out + (size_t)(tm * 16 + half * 8) * D_FEAT + tn * 16 + l;
#pragma unroll
  for (int v = 0; v < 8; ++v) {
    const float r = c[v] > 0.f ? c[v] : 0.f;            // fused ReLU
    O[(size_t)v * D_FEAT] = r;
  }
}

// ---------------------------------------------------------------------- launch
extern "C" void kernel_launch(void* const* d_in, const int* in_sizes, int n_in,
                              void* d_out, int out_size, void* d_ws, size_t ws_size,
                              hipStream_t stream) {
  const float* x     = (const float*)d_in[0];
  const int*   erow  = (const int*)d_in[1];
  const int*   ecol  = (const int*)d_in[2];
  const float* eval  = (const float*)d_in[3];
  const float* weight= (const float*)d_in[4];
  float*       out   = (float*)d_out;

  const int M = in_sizes[0] / D_FEAT;   // 100000 nodes
  const int E = in_sizes[1];            // 3.2M edges

  float* S = (float*)d_ws;              // support scratch: M*256 floats (102.4 MB)

  // 1) zero the accumulation buffer (required every call: atomics accumulate)
  {
    const long n4 = (long)M * (D_FEAT / 4);
    const int blocks = 4096;
    lpf_zero_f4<<<blocks, 256, 0, stream>>>((float4*)S, n4);
  }

  // 2) SpMM scatter: one wave per edge, grid-stride
  {
    const int blocks = 8192;            // 8192*8 = 65536 waves in flight
    lpf_spmm_scatter<<<blocks, 256, 0, stream>>>(x, erow, ecol, eval, S, E);
  }

  // 3) fp32 WMMA GEMM + fused ReLU
  {
    const int tilesM = (M + 15) / 16;
    const int tilesN = D_FEAT / 16;
    const int tiles = tilesM * tilesN;  // 100000
    const int wavesPerBlock = 8;        // 256 threads
    const int blocks = (tiles + wavesPerBlock - 1) / wavesPerBlock;
    lpf_gemm_relu_wmma<<<blocks, 256, 0, stream>>>(S, weight, out, M);
  }
}